// MultiHeadAttention_64046552318204
// MI455X (gfx1250) — compile-verified
//
#include <hip/hip_runtime.h>

// MI455X / gfx1250, wave32. All matmuls via V_WMMA_F32_16X16X4_F32 (fp32-exact).
// WMMA fragment layouts per CDNA5 ISA:
//   A(16x4): element k = 2*(lane>=16) + vgpr, row m = lane&15  -> aligned float2 per lane
//   B(4x16): element k = 2*(lane>=16) + vgpr, col n = lane&15
//   C/D(16x16): row = vgpr + 8*(lane>=16), col = lane&15

typedef __attribute__((ext_vector_type(2))) float v2f;
typedef __attribute__((ext_vector_type(8))) float v8f;

__device__ __forceinline__ v8f wmma4(v2f a, v2f b, v8f c) {
  return __builtin_amdgcn_wmma_f32_16x16x4_f32(
      /*neg_a=*/false, a, /*neg_b=*/false, b,
      /*c_mod=*/(short)0, c, /*reuse_a=*/false, /*reuse_b=*/false);
}

__device__ __forceinline__ float redmax16(float x) {
#pragma unroll
  for (int o = 8; o > 0; o >>= 1) x = fmaxf(x, __shfl_xor(x, o, 32));
  return x;
}
__device__ __forceinline__ float redsum16(float x) {
#pragma unroll
  for (int o = 8; o > 0; o >>= 1) x += __shfl_xor(x, o, 32);
  return x;
}

// ---------------------------------------------------------------------------
// Kernel 1: QKV projection. Q/K/V[b,n,l,h] = sum_m x[b,l,m] * W[n,m,h]
// GEMM view: [4096,1024] x [1024,1024]; each wave computes a 16x64 strip
// (one head's full H range) -> A fragment reused across 4 column tiles.
// grid (256 row-tiles, 16 heads, 3 {q,k,v}), block 32 (one wave).
// ---------------------------------------------------------------------------
__global__ __launch_bounds__(32) void qkv_proj_kernel(
    const float* __restrict__ x, const float* __restrict__ wq,
    const float* __restrict__ wk, const float* __restrict__ wv,
    float* __restrict__ Q, float* __restrict__ K, float* __restrict__ V) {
  const int lane = threadIdx.x;
  const int half = lane >> 4;   // k-group selector
  const int m = lane & 15;      // A row / B col within tile
  const int row0 = blockIdx.x * 16;
  const int nh = blockIdx.y;    // head index == column strip / 64
  const int sel = blockIdx.z;
  const float* W = (sel == 0) ? wq : ((sel == 1) ? wk : wv);
  float* Dst = (sel == 0) ? Q : ((sel == 1) ? K : V);

  const float* ap = x + (size_t)(row0 + m) * 1024;
  const float* wp = W + (size_t)nh * (1024 * 64) + m;  // + k*64 + ct*16

  v8f c0 = {}, c1 = {}, c2 = {}, c3 = {};
#pragma unroll 4
  for (int kt = 0; kt < 256; ++kt) {
    const int k = kt * 4 + 2 * half;
    v2f a = *(const v2f*)(ap + k);
    const float* w0 = wp + (size_t)k * 64;
    const float* w1 = wp + (size_t)(k + 1) * 64;
    v2f b0, b1, b2, b3;
    b0.x = w0[0];  b0.y = w1[0];
    b1.x = w0[16]; b1.y = w1[16];
    b2.x = w0[32]; b2.y = w1[32];
    b3.x = w0[48]; b3.y = w1[48];
    c0 = wmma4(a, b0, c0);
    c1 = wmma4(a, b1, c1);
    c2 = wmma4(a, b2, c2);
    c3 = wmma4(a, b3, c3);
  }
#pragma unroll
  for (int v = 0; v < 8; ++v) {
    const int grow = row0 + half * 8 + v;  // global row in [0,4096)
    const int bi = grow >> 10, li = grow & 1023;
    float* dp = Dst + (((size_t)bi * 16 + nh) * 1024 + li) * 64 + m;
    dp[0] = c0[v];
    dp[16] = c1[v];
    dp[32] = c2[v];
    dp[48] = c3[v];
  }
}

// ---------------------------------------------------------------------------
// Kernel 2: relative-position bias tables.
// QB[b,n,l,r] = Q[b,n,l,:] . q_tab[r,:]   (r in [0,127), padded stride 128)
// KB[b,n,l,r] = K[b,n,l,:] . k_tab[r,:]
// grid (64 l-tiles, 8 r-tiles, 128 = bn*2+s), block 32.
// ---------------------------------------------------------------------------
__global__ __launch_bounds__(32) void embed_bias_kernel(
    const float* __restrict__ Q, const float* __restrict__ K,
    const float* __restrict__ qtab, const float* __restrict__ ktab,
    float* __restrict__ QB, float* __restrict__ KB) {
  const int lane = threadIdx.x, half = lane >> 4, m = lane & 15;
  const int l0 = blockIdx.x * 16, r0 = blockIdx.y * 16;
  const int z = blockIdx.z;
  const int bn = z >> 1, s = z & 1;
  const float* A = ((s == 0) ? Q : K) + ((size_t)bn * 1024 + l0 + m) * 64;
  const float* T = (s == 0) ? qtab : ktab;
  float* Dst = (s == 0) ? QB : KB;
  const int col = r0 + m;
  const float* tp = T + (size_t)min(col, 126) * 64;  // clamp pad column

  v8f c = {};
#pragma unroll
  for (int t = 0; t < 16; ++t) {
    const int k = t * 4 + 2 * half;
    v2f a = *(const v2f*)(A + k);
    v2f bb = *(const v2f*)(tp + k);
    c = wmma4(a, bb, c);
  }
#pragma unroll
  for (int v = 0; v < 8; ++v) {
    const int row = half * 8 + v;
    Dst[((size_t)bn * 1024 + l0 + row) * 128 + col] = c[v];
  }
}

// ---------------------------------------------------------------------------
// Kernel 3: fused attention for one (b,n, 16 query rows).
// Two-pass exact softmax; value-side relative bias via pe-binning + WMMA.
// grid (64 i-tiles, 64 bn), block 32, LDS = 1KB ptile + 8KB pe.
// ---------------------------------------------------------------------------
__global__ __launch_bounds__(32) void attn_kernel(
    const float* __restrict__ Q, const float* __restrict__ K,
    const float* __restrict__ V, const float* __restrict__ QB,
    const float* __restrict__ KB, const int* __restrict__ mask,
    const float* __restrict__ vtab, float* __restrict__ O2) {
  __shared__ float ptile[16 * 16];
  __shared__ float pe[16 * 128];
  const int lane = threadIdx.x, half = lane >> 4, m = lane & 15;
  const int i0 = blockIdx.x * 16;
  const int bn = blockIdx.y;
  const int b = bn >> 4, n = bn & 15;
  const float* Qb = Q + (size_t)bn * 1024 * 64;
  const float* Kb = K + (size_t)bn * 1024 * 64;
  const float* Vb = V + (size_t)bn * 1024 * 64;
  const float* QBb = QB + (size_t)bn * 1024 * 128;
  const float* KBb = KB + (size_t)bn * 1024 * 128;
  const int* mb = mask + b * 1024;

  for (int t = lane; t < 16 * 128; t += 32) pe[t] = 0.f;
  __syncthreads();

  // Preload A fragments of the Q tile (16x64): k = 4t + 2*half {+0,+1}
  v2f qf[16];
  const float* qp = Qb + (size_t)(i0 + m) * 64;
#pragma unroll
  for (int t = 0; t < 16; ++t) qf[t] = *(const v2f*)(qp + t * 4 + 2 * half);

  float runm[8], runl[8];
#pragma unroll
  for (int v = 0; v < 8; ++v) { runm[v] = -3.0e38f; runl[v] = 0.f; }

  // ---- pass 1: exact row max and exp-sum ----
  for (int jt = 0; jt < 64; ++jt) {
    const int j0 = jt * 16;
    const float* kp = Kb + (size_t)(j0 + m) * 64;
    v8f s = {};
#pragma unroll
    for (int t = 0; t < 16; ++t) {
      v2f kv = *(const v2f*)(kp + t * 4 + 2 * half);
      s = wmma4(qf[t], kv, s);
    }
    const int j = j0 + m;
    const int mk = mb[j];
#pragma unroll
    for (int v = 0; v < 8; ++v) {
      const int i = i0 + half * 8 + v;
      const int d = min(63, max(-63, i - j)) + 63;
      float sv = (s[v] + KBb[(size_t)j * 128 + d] + QBb[(size_t)i * 128 + d]) * 0.125f;
      sv = (mk == 0) ? -1e9f : sv;
      const float tm = redmax16(sv);
      const float nm = fmaxf(runm[v], tm);
      const float ts = redsum16(__expf(sv - nm));
      runl[v] = runl[v] * __expf(runm[v] - nm) + ts;
      runm[v] = nm;
    }
  }
  float invl[8];
#pragma unroll
  for (int v = 0; v < 8; ++v) invl[v] = 1.0f / runl[v];

  v8f oacc[4] = {};
  float pe0[8] = {}, pe126[8] = {};  // clipped-bin partials (register-resident)

  // ---- pass 2: normalized P, PV accumulation, pe binning ----
  for (int jt = 0; jt < 64; ++jt) {
    const int j0 = jt * 16;
    const float* kp = Kb + (size_t)(j0 + m) * 64;
    v8f s = {};
#pragma unroll
    for (int t = 0; t < 16; ++t) {
      v2f kv = *(const v2f*)(kp + t * 4 + 2 * half);
      s = wmma4(qf[t], kv, s);
    }
    const int j = j0 + m;
    const int mk = mb[j];
#pragma unroll
    for (int v = 0; v < 8; ++v) {
      const int row = half * 8 + v;
      const int i = i0 + row;
      const int d = min(63, max(-63, i - j)) + 63;
      float sv = (s[v] + KBb[(size_t)j * 128 + d] + QBb[(size_t)i * 128 + d]) * 0.125f;
      sv = (mk == 0) ? -1e9f : sv;
      const float p = __expf(sv - runm[v]) * invl[v];
      ptile[row * 16 + m] = p;
      // For 0<d<126 each (row,d) is hit by exactly one lane -> race-free.
      if (d == 0) pe0[v] += p;
      else if (d == 126) pe126[v] += p;
      else pe[row * 128 + d] += p;
    }
    __syncthreads();
#pragma unroll
    for (int ks = 0; ks < 4; ++ks) {
      v2f a = *(const v2f*)&ptile[m * 16 + ks * 4 + 2 * half];
      const float* vp = Vb + (size_t)(j0 + ks * 4 + 2 * half) * 64;
#pragma unroll
      for (int ct = 0; ct < 4; ++ct) {
        const int h = ct * 16 + m;
        v2f bb;
        bb.x = vp[h];
        bb.y = vp[64 + h];
        oacc[ct] = wmma4(a, bb, oacc[ct]);
      }
    }
    __syncthreads();
  }

  // fold clipped bins into pe
#pragma unroll
  for (int v = 0; v < 8; ++v) {
    const float s0 = redsum16(pe0[v]);
    const float s126 = redsum16(pe126[v]);
    if (m == 0) {
      const int row = half * 8 + v;
      pe[row * 128 + 0] += s0;
      pe[row * 128 + 126] += s126;
    }
  }
  __syncthreads();

  // out += pe(16x128) @ v_tab(128x64); pe[:,127]==0, clamp table row for pad.
#pragma unroll 2
  for (int kt = 0; kt < 32; ++kt) {
    const int k = kt * 4 + 2 * half;
    v2f a = *(const v2f*)&pe[m * 128 + k];
    const float* b0 = vtab + (size_t)min(k, 126) * 64;
    const float* b1 = vtab + (size_t)min(k + 1, 126) * 64;
#pragma unroll
    for (int ct = 0; ct < 4; ++ct) {
      const int h = ct * 16 + m;
      v2f bb;
      bb.x = b0[h];
      bb.y = b1[h];
      oacc[ct] = wmma4(a, bb, oacc[ct]);
    }
  }

  // store concat-head layout O2[b, i, n*64+h]
#pragma unroll
  for (int ct = 0; ct < 4; ++ct) {
#pragma unroll
    for (int v = 0; v < 8; ++v) {
      const int row = half * 8 + v;
      O2[((size_t)b * 1024 + i0 + row) * 1024 + n * 64 + ct * 16 + m] = oacc[ct][v];
    }
  }
}

// ---------------------------------------------------------------------------
// Kernel 4: output projection. Y[4096,1024] = O2[4096,1024] @ w_o(flat 1024x1024)
// (w_o's [N,H,D] layout flattens exactly to the needed [n*64+h, m] operand.)
// Each wave computes a 16x64 strip: A fragment reused across 4 column tiles.
// grid (256 row-tiles, 16 col-strips), block 32.
// ---------------------------------------------------------------------------
__global__ __launch_bounds__(32) void out_proj_kernel(
    const float* __restrict__ O2, const float* __restrict__ wo,
    float* __restrict__ Y) {
  const int lane = threadIdx.x, half = lane >> 4, m = lane & 15;
  const int row0 = blockIdx.x * 16, col0 = blockIdx.y * 64;
  const float* ap = O2 + (size_t)(row0 + m) * 1024;
  const float* bp = wo + col0 + m;
  v8f c0 = {}, c1 = {}, c2 = {}, c3 = {};
#pragma unroll 4
  for (int kt = 0; kt < 256; ++kt) {
    const int k = kt * 4 + 2 * half;
    v2f a = *(const v2f*)(ap + k);
    const float* w0 = bp + (size_t)k * 1024;
    const float* w1 = bp + (size_t)(k + 1) * 1024;
    v2f b0, b1, b2, b3;
    b0.x = w0[0];  b0.y = w1[0];
    b1.x = w0[16]; b1.y = w1[16];
    b2.x = w0[32]; b2.y = w1[32];
    b3.x = w0[48]; b3.y = w1[48];
    c0 = wmma4(a, b0, c0);
    c1 = wmma4(a, b1, c1);
    c2 = wmma4(a, b2, c2);
    c3 = wmma4(a, b3, c3);
  }
#pragma unroll
  for (int v = 0; v < 8; ++v) {
    float* yp = Y + (size_t)(row0 + half * 8 + v) * 1024 + col0 + m;
    yp[0] = c0[v];
    yp[16] = c1[v];
    yp[32] = c2[v];
    yp[48] = c3[v];
  }
}

extern "C" void kernel_launch(void* const* d_in, const int* in_sizes, int n_in,
                              void* d_out, int out_size, void* d_ws, size_t ws_size,
                              hipStream_t stream) {
  const float* x = (const float*)d_in[0];
  const int* mask = (const int*)d_in[1];
  const float* wq = (const float*)d_in[2];
  const float* wk = (const float*)d_in[3];
  const float* wv = (const float*)d_in[4];
  const float* wo = (const float*)d_in[5];
  const float* qtab = (const float*)d_in[6];
  const float* ktab = (const float*)d_in[7];
  const float* vtab = (const float*)d_in[8];
  float* Y = (float*)d_out;

  float* ws = (float*)d_ws;
  float* Q = ws;                    // [4,16,1024,64]
  float* K = Q + 4194304;           // [4,16,1024,64]
  float* V = K + 4194304;           // [4,16,1024,64]
  float* QB = V + 4194304;          // [4,16,1024,128]
  float* KB = QB + 8388608;         // [4,16,1024,128]
  float* O2 = KB + 8388608;         // [4,1024,1024]

  qkv_proj_kernel<<<dim3(256, 16, 3), 32, 0, stream>>>(x, wq, wk, wv, Q, K, V);
  embed_bias_kernel<<<dim3(64, 8, 128), 32, 0, stream>>>(Q, K, qtab, ktab, QB, KB);
  attn_kernel<<<dim3(64, 64), 32, 0, stream>>>(Q, K, V, QB, KB, mask, vtab, O2);
  out_proj_kernel<<<dim3(256, 16), 32, 0, stream>>>(O2, wo, Y);
}